// GrassmannConditional_33990371181115
// MI455X (gfx1250) — compile-verified
//
#include <hip/hip_runtime.h>
#include <hip/hip_bf16.h>

typedef float v2f __attribute__((ext_vector_type(2)));
typedef float v8f __attribute__((ext_vector_type(8)));

#define B_    8192
#define D_    32
#define K_    8
#define H_    512
#define CTX_  64
#define NBC   16384   /* K_*2*D_*D_ */
#define EPS_  1e-4f

static __device__ __forceinline__ v8f wmma_f32(v2f a, v2f b, v8f c) {
  // D = A(16x4,f32) * B(4x16,f32) + C(16x16,f32)
  return __builtin_amdgcn_wmma_f32_16x16x4_f32(false, a, false, b, (short)0, c, false, false);
}

// ---------------------------------------------------------------------------
// K1: h = relu(context @ W_h + b_h)   [8192,64]x[64,512]
// block = 256 (8 waves), tile 16 rows x 128 cols; grid (512, 4)
// ---------------------------------------------------------------------------
__global__ __launch_bounds__(256) void k_hidden(const float* __restrict__ ctx,
                                                const float* __restrict__ W_h,
                                                const float* __restrict__ b_h,
                                                float* __restrict__ h) {
  const int lane = threadIdx.x & 31;
  const int w    = threadIdx.x >> 5;
  const int hi   = lane >> 4;      // 0/1 : K-half of the fragment
  const int nl   = lane & 15;      // column within tile / A row
  const int b0   = blockIdx.x * 16;
  const int n0   = blockIdx.y * 128 + w * 16;

  const float* arow = ctx + (size_t)(b0 + nl) * CTX_;
  v8f acc;
#pragma unroll
  for (int r = 0; r < 8; ++r) acc[r] = 0.f;

  for (int k0 = 0; k0 < CTX_; k0 += 4) {
    const int ka = k0 + 2 * hi;
    v2f a; a[0] = arow[ka]; a[1] = arow[ka + 1];
    v2f b; b[0] = W_h[(size_t)ka * H_ + n0 + nl];
           b[1] = W_h[(size_t)(ka + 1) * H_ + n0 + nl];
    acc = wmma_f32(a, b, acc);
  }
  const float bias = b_h[n0 + nl];
#pragma unroll
  for (int r = 0; r < 8; ++r) {
    const int m = r + 8 * hi;
    const float v = acc[r] + bias;
    h[(size_t)(b0 + m) * H_ + n0 + nl] = v > 0.f ? v : 0.f;
  }
}

// ---------------------------------------------------------------------------
// K2: mixing_p = softmax(h @ W_logit + b_logit), one wave per sample
// ---------------------------------------------------------------------------
__global__ __launch_bounds__(256) void k_mix(const float* __restrict__ h,
                                             const float* __restrict__ W_logit,
                                             const float* __restrict__ b_logit,
                                             float* __restrict__ mix) {
  const int lane = threadIdx.x & 31;
  const int w    = threadIdx.x >> 5;
  const int b    = blockIdx.x * 8 + w;

  float acc[8];
#pragma unroll
  for (int j = 0; j < 8; ++j) acc[j] = 0.f;

  for (int c = lane; c < H_; c += 32) {
    const float hv = h[(size_t)b * H_ + c];
    const float4 w0 = *(const float4*)&W_logit[c * 8];
    const float4 w1 = *(const float4*)&W_logit[c * 8 + 4];
    acc[0] += hv * w0.x; acc[1] += hv * w0.y; acc[2] += hv * w0.z; acc[3] += hv * w0.w;
    acc[4] += hv * w1.x; acc[5] += hv * w1.y; acc[6] += hv * w1.z; acc[7] += hv * w1.w;
  }
#pragma unroll
  for (int j = 0; j < 8; ++j) {
#pragma unroll
    for (int off = 16; off > 0; off >>= 1)
      acc[j] += __shfl_down(acc[j], off, 32);
  }
  if (lane == 0) {
    float m = -1e30f;
#pragma unroll
    for (int j = 0; j < 8; ++j) { acc[j] += b_logit[j]; m = fmaxf(m, acc[j]); }
    float sum = 0.f;
#pragma unroll
    for (int j = 0; j < 8; ++j) { acc[j] = __expf(acc[j] - m); sum += acc[j]; }
    const float inv = 1.f / sum;
#pragma unroll
    for (int j = 0; j < 8; ++j) mix[(size_t)b * K_ + j] = acc[j] * inv;
  }
}

// ---------------------------------------------------------------------------
// K3 helpers
// ---------------------------------------------------------------------------
// One WMMA GEMM pass: [16,512] @ [512,1024] slice of W_bc -> LDS bc[16][1024].
// K-outer / tile-inner: the A fragment is loaded once per K-step and reused
// by 4 back-to-back WMMAs (4 live accumulators).
__device__ __forceinline__ void gemm_pass(const float* __restrict__ h,
                                          const float* __restrict__ W_bc,
                                          const float* __restrict__ b_bc,
                                          float* bcL, int b0, int k, int s,
                                          int w, int lane) {
  const int hi = lane >> 4, nl = lane & 15;
  const float* arow  = h + (size_t)(b0 + nl) * H_;
  const int    cbase = k * 2048 + s * 1024 + w * 64 + nl;  // tile tt at +16*tt

  v8f acc[4];
#pragma unroll
  for (int tt = 0; tt < 4; ++tt)
#pragma unroll
    for (int r = 0; r < 8; ++r) acc[tt][r] = 0.f;

  for (int k0 = 0; k0 < H_; k0 += 4) {
    const int ka = k0 + 2 * hi;
    v2f a; a[0] = arow[ka]; a[1] = arow[ka + 1];
    const float* w0 = W_bc + (size_t)ka * NBC + cbase;
#pragma unroll
    for (int tt = 0; tt < 4; ++tt) {
      v2f b; b[0] = w0[tt * 16];
             b[1] = w0[NBC + tt * 16];
      acc[tt] = wmma_f32(a, b, acc[tt]);
    }
  }
#pragma unroll
  for (int tt = 0; tt < 4; ++tt) {
    const int n0   = w * 64 + tt * 16;
    const float bias = b_bc[cbase + tt * 16];
#pragma unroll
    for (int r = 0; r < 8; ++r)
      bcL[(r + 8 * hi) * 1024 + n0 + nl] = acc[tt][r] + bias;
  }
}

// Gram matrix: g[l] = sum_j M[lane][j]*M[l][j] + eps*I, M = 32x32 in LDS chunk
__device__ __forceinline__ void gram_from_lds(const float* chunk, int lane,
                                              float (&g)[32]) {
  float4 rv[8];
  const float4* own = (const float4*)(chunk + lane * 32);
#pragma unroll
  for (int j = 0; j < 8; ++j) rv[j] = own[j];
#pragma unroll
  for (int l = 0; l < 32; ++l) {
    const float4* qr = (const float4*)(chunk + l * 32);
    float a = 0.f;
#pragma unroll
    for (int j = 0; j < 8; ++j) {
      const float4 q = qr[j];
      a += rv[j].x * q.x + rv[j].y * q.y + rv[j].z * q.z + rv[j].w * q.w;
    }
    g[l] = a + ((l == lane) ? EPS_ : 0.f);
  }
}

// ---------------------------------------------------------------------------
// K3: fused BC-GEMM + Gram + log-det ratio.  det(M) = det(A)/det(S) where
// A = rows of (BBt+eps I) where x=1 else rows of (CCt+eps I); S = sum.
// block = 512 thr (16 waves) = 16 batch rows, one k.  grid (512, 8).
// ---------------------------------------------------------------------------
__global__ __launch_bounds__(512) void k_comp(const float* __restrict__ h,
                                              const float* __restrict__ W_bc,
                                              const float* __restrict__ b_bc,
                                              const float* __restrict__ x,
                                              float* __restrict__ comp) {
  __shared__ float bc[16 * 1024];   // 64 KB

  const int lane = threadIdx.x & 31;
  const int w    = threadIdx.x >> 5;     // 0..15, also the sample this wave owns
  const int b0   = blockIdx.x * 16;
  const int k    = blockIdx.y;

  float bbt[32], cct[32];

  // pass s=0 : Bm -> LDS, then BBt in registers (row = lane)
  gemm_pass(h, W_bc, b_bc, bc, b0, k, 0, w, lane);
  __syncthreads();
  gram_from_lds(bc + w * 1024, lane, bbt);
  __syncthreads();
  // pass s=1 : Cm -> LDS, then CCt
  gemm_pass(h, W_bc, b_bc, bc, b0, k, 1, w, lane);
  __syncthreads();
  gram_from_lds(bc + w * 1024, lane, cct);

  // Build A (row-select by x) and S (sum), row-per-lane.
  const float xv  = x[(size_t)(b0 + w) * D_ + lane];
  const bool  sel = xv > 0.5f;
  float arw[32], srw[32];
#pragma unroll
  for (int l = 0; l < 32; ++l) {
    srw[l] = bbt[l] + cct[l];
    arw[l] = sel ? bbt[l] : cct[l];
  }

  // --- log det(S): Gaussian elimination, no pivoting (SPD) ---
  float logS = 0.f;
#pragma unroll
  for (int t = 0; t < 32; ++t) {
    float piv = __shfl(srw[t], t, 32);
    piv = fmaxf(piv, 1e-30f);
    logS += __logf(piv);
    const float f = srw[t] / piv;
#pragma unroll
    for (int j = t + 1; j < 32; ++j) {
      const float pj = __shfl(srw[j], t, 32);
      srw[j] = (lane > t) ? (srw[j] - f * pj) : srw[j];
    }
  }

  // --- log|det(A)| + sign: partial pivoting via lane argmax + row shuffle ---
  float logA = 0.f, signA = 1.f;
  int zero = 0;
#pragma unroll
  for (int t = 0; t < 32; ++t) {
    float av  = (lane >= t) ? fabsf(arw[t]) : -1.f;
    int  pidx = lane;
#pragma unroll
    for (int off = 16; off > 0; off >>= 1) {
      const float ov = __shfl_xor(av, off, 32);
      const int   oi = __shfl_xor(pidx, off, 32);
      if (ov > av || (ov == av && oi < pidx)) { av = ov; pidx = oi; }
    }
    // swap rows t <-> pidx (branchless lane remap)
    const int src = (lane == t) ? pidx : ((lane == pidx) ? t : lane);
#pragma unroll
    for (int j = 0; j < 32; ++j) arw[j] = __shfl(arw[j], src, 32);
    signA = (pidx != t) ? -signA : signA;

    const float piv  = __shfl(arw[t], t, 32);
    const float apiv = fabsf(piv);
    zero |= (apiv < 1e-30f) ? 1 : 0;
    signA = (piv < 0.f) ? -signA : signA;
    logA += __logf(fmaxf(apiv, 1e-30f));
    const float pv = (apiv < 1e-30f) ? 1.f : piv;
    const float f  = arw[t] / pv;
#pragma unroll
    for (int j = t + 1; j < 32; ++j) {
      const float pj = __shfl(arw[j], t, 32);
      arw[j] = (lane > t) ? (arw[j] - f * pj) : arw[j];
    }
  }

  const float c = zero ? 0.f : signA * __expf(logA - logS);
  if (lane == 0) comp[(size_t)(b0 + w) * K_ + k] = c;
}

// ---------------------------------------------------------------------------
// K4: out[b] = sum_k mix[b,k]*comp[b,k]
// ---------------------------------------------------------------------------
__global__ __launch_bounds__(256) void k_out(const float* __restrict__ mix,
                                             const float* __restrict__ comp,
                                             float* __restrict__ out) {
  const int b = blockIdx.x * 256 + threadIdx.x;
  float s = 0.f;
#pragma unroll
  for (int j = 0; j < 8; ++j)
    s += mix[(size_t)b * K_ + j] * comp[(size_t)b * K_ + j];
  out[b] = s;
}

// ---------------------------------------------------------------------------
extern "C" void kernel_launch(void* const* d_in, const int* in_sizes, int n_in,
                              void* d_out, int out_size, void* d_ws, size_t ws_size,
                              hipStream_t stream) {
  (void)in_sizes; (void)n_in; (void)out_size; (void)ws_size;
  const float* inputs  = (const float*)d_in[0];
  const float* context = (const float*)d_in[1];
  const float* W_h     = (const float*)d_in[2];
  const float* b_h     = (const float*)d_in[3];
  const float* W_logit = (const float*)d_in[4];
  const float* b_logit = (const float*)d_in[5];
  const float* W_bc    = (const float*)d_in[6];
  const float* b_bc    = (const float*)d_in[7];
  float* out = (float*)d_out;

  float* h    = (float*)d_ws;                 // 8192*512 floats = 16 MB
  float* mix  = h + (size_t)B_ * H_;          // 8192*8
  float* comp = mix + (size_t)B_ * K_;        // 8192*8

  k_hidden<<<dim3(B_ / 16, H_ / 128), 256, 0, stream>>>(context, W_h, b_h, h);
  k_mix   <<<dim3(B_ / 8),            256, 0, stream>>>(h, W_logit, b_logit, mix);
  k_comp  <<<dim3(B_ / 16, K_),       512, 0, stream>>>(h, W_bc, b_bc, inputs, comp);
  k_out   <<<dim3(B_ / 256),          256, 0, stream>>>(mix, comp, out);
}